// HGNN_82463372083254
// MI455X (gfx1250) — compile-verified
//
#include <hip/hip_runtime.h>
#include <hip/hip_bf16.h>

#define N_NODES   100000
#define N_EDGES   20000
#define NNZ_CNT   800000
#define D_IN      128
#define D         256
#define EPS_F     1e-5f

typedef float v2f __attribute__((ext_vector_type(2)));
typedef float v8f __attribute__((ext_vector_type(8)));

__device__ __forceinline__ void atomAddF(float* p, float v) {
    __hip_atomic_fetch_add(p, v, __ATOMIC_RELAXED, __HIP_MEMORY_SCOPE_AGENT);
}

// ---------------------------------------------------------------------------
// Degree / edge-cardinality accumulation (float, matches reference segment_sum of ones)
// ---------------------------------------------------------------------------
__global__ __launch_bounds__(256) void degcnt_kernel(const int* __restrict__ node_idx,
                                                     const int* __restrict__ edge_idx,
                                                     float* __restrict__ deg,
                                                     float* __restrict__ cnt) {
    int i = blockIdx.x * 256 + threadIdx.x;
    if (i < NNZ_CNT) {
        atomAddF(&deg[node_idx[i]], 1.0f);
        atomAddF(&cnt[edge_idx[i]], 1.0f);
    }
}

// ---------------------------------------------------------------------------
// WMMA GEMM:  C[M, 256] = X[M, K] @ W[256, K]^T   using V_WMMA_F32_16X16X4_F32
// One wave computes a 16(M) x 64(N) strip with 4 accumulators.
// Launch: total waves = (M/16) * 4, blockDim = 256 (8 waves/block).
// ---------------------------------------------------------------------------
template<int K>
__global__ __launch_bounds__(256) void gemm_wmma_kernel(const float* __restrict__ X,
                                                        const float* __restrict__ W,
                                                        float* __restrict__ C,
                                                        int mTiles) {
    const int wid  = (blockIdx.x * 256 + threadIdx.x) >> 5;   // global wave id
    const int lane = threadIdx.x & 31;
    const int mT   = wid >> 2;
    const int nT   = wid & 3;
    if (mT >= mTiles) return;                                 // wave-uniform
    const int m0   = mT * 16;
    const int n0   = nT * 64;
    const int half = lane >> 4;        // 0: K pair {k0,k0+1}, 1: {k0+2,k0+3}
    const int l16  = lane & 15;

    const float* __restrict__ xrow = X + (size_t)(m0 + l16) * K;
    const float* __restrict__ wrow = W + (size_t)(n0 + l16) * K;

    v8f acc0 = {}, acc1 = {}, acc2 = {}, acc3 = {};

    #pragma unroll 8
    for (int k0 = 0; k0 < K; k0 += 4) {
        const int ko = k0 + 2 * half;
        v2f a  = *reinterpret_cast<const v2f*>(xrow + ko);
        v2f b0 = *reinterpret_cast<const v2f*>(wrow + (size_t)0 * 16 * K + ko);
        v2f b1 = *reinterpret_cast<const v2f*>(wrow + (size_t)1 * 16 * K + ko);
        v2f b2 = *reinterpret_cast<const v2f*>(wrow + (size_t)2 * 16 * K + ko);
        v2f b3 = *reinterpret_cast<const v2f*>(wrow + (size_t)3 * 16 * K + ko);
        acc0 = __builtin_amdgcn_wmma_f32_16x16x4_f32(false, a, false, b0, (short)0, acc0, false, false);
        acc1 = __builtin_amdgcn_wmma_f32_16x16x4_f32(false, a, false, b1, (short)0, acc1, false, false);
        acc2 = __builtin_amdgcn_wmma_f32_16x16x4_f32(false, a, false, b2, (short)0, acc2, false, false);
        acc3 = __builtin_amdgcn_wmma_f32_16x16x4_f32(false, a, false, b3, (short)0, acc3, false, false);
    }

    float* __restrict__ ccol = C + n0 + l16;
    #pragma unroll
    for (int r = 0; r < 8; ++r) {
        const size_t m = (size_t)(m0 + r + 8 * half);
        ccol[m * D + 0 * 16] = acc0[r];
        ccol[m * D + 1 * 16] = acc1[r];
        ccol[m * D + 2 * 16] = acc2[r];
        ccol[m * D + 3 * 16] = acc3[r];
    }
}

// ---------------------------------------------------------------------------
// Scatter-add of full 256-wide rows:  dst[dst_idx[i]] += src[src_idx[i]]
// One block (256 threads) per nnz entry; coalesced load + coalesced FP atomic.
// ---------------------------------------------------------------------------
__global__ __launch_bounds__(256) void scatter_rows_kernel(const float* __restrict__ src,
                                                           const int* __restrict__ src_idx,
                                                           const int* __restrict__ dst_idx,
                                                           float* __restrict__ dst) {
    const int i = blockIdx.x;
    const int f = threadIdx.x;
    const int s = src_idx[i];
    const int d = dst_idx[i];
    const float v = src[(size_t)s * D + f];
    atomAddF(&dst[(size_t)d * D + f], v);
}

// ---------------------------------------------------------------------------
// Scale hyperedge features by B^{-1}
// ---------------------------------------------------------------------------
__global__ __launch_bounds__(256) void scale_edges_kernel(float* __restrict__ E,
                                                          const float* __restrict__ cnt) {
    const int idx = blockIdx.x * 256 + threadIdx.x;
    if (idx < N_EDGES * D) {
        const float c = cnt[idx >> 8];
        const float inv = (c > 0.0f) ? 1.0f / c : 0.0f;
        E[idx] *= inv;
    }
}

// ---------------------------------------------------------------------------
// Fused: out = PReLU(LayerNorm(acc * D^{-1} + bias))   one row per block
// ---------------------------------------------------------------------------
__global__ __launch_bounds__(256) void fused_scale_ln_prelu_kernel(const float* __restrict__ acc,
                                                                   const float* __restrict__ deg,
                                                                   const float* __restrict__ bias,
                                                                   const float* __restrict__ g,
                                                                   const float* __restrict__ beta,
                                                                   const float* __restrict__ pa,
                                                                   float* __restrict__ out) {
    __shared__ float sred[256];
    const int r = blockIdx.x;
    const int f = threadIdx.x;
    const float dg   = deg[r];
    const float dinv = (dg > 0.0f) ? 1.0f / dg : 0.0f;
    const float v = acc[(size_t)r * D + f] * dinv + bias[f];

    sred[f] = v;
    __syncthreads();
    #pragma unroll
    for (int s = 128; s > 0; s >>= 1) { if (f < s) sred[f] += sred[f + s]; __syncthreads(); }
    const float mu = sred[0] * (1.0f / D);
    __syncthreads();

    const float dv = v - mu;
    sred[f] = dv * dv;
    __syncthreads();
    #pragma unroll
    for (int s = 128; s > 0; s >>= 1) { if (f < s) sred[f] += sred[f + s]; __syncthreads(); }
    const float var = sred[0] * (1.0f / D);   // population variance (ddof=0)

    const float y = dv * rsqrtf(var + EPS_F) * g[f] + beta[f];
    const float a = pa[0];
    out[(size_t)r * D + f] = (y >= 0.0f) ? y : a * y;
}

// ---------------------------------------------------------------------------
// Weight standardization (ddof=1, per row of W2[256,256])
// ---------------------------------------------------------------------------
__global__ __launch_bounds__(256) void wstd_kernel(const float* __restrict__ W,
                                                   float* __restrict__ Ws) {
    __shared__ float sred[256];
    const int r = blockIdx.x;
    const int f = threadIdx.x;
    const float w = W[(size_t)r * D + f];

    sred[f] = w;
    __syncthreads();
    #pragma unroll
    for (int s = 128; s > 0; s >>= 1) { if (f < s) sred[f] += sred[f + s]; __syncthreads(); }
    const float mean = sred[0] * (1.0f / D);
    __syncthreads();

    const float dv = w - mean;
    sred[f] = dv * dv;
    __syncthreads();
    #pragma unroll
    for (int s = 128; s > 0; s >>= 1) { if (f < s) sred[f] += sred[f + s]; __syncthreads(); }
    const float var = sred[0] * (1.0f / (D - 1));   // unbiased (ddof=1)

    Ws[(size_t)r * D + f] = dv * rsqrtf(var + EPS_F);
}

// ---------------------------------------------------------------------------
extern "C" void kernel_launch(void* const* d_in, const int* in_sizes, int n_in,
                              void* d_out, int out_size, void* d_ws, size_t ws_size,
                              hipStream_t stream) {
    const float* edge_fea = (const float*)d_in[0];   // [100000,128]
    const int*   node_idx = (const int*)  d_in[1];   // [800000]
    const int*   edge_idx = (const int*)  d_in[2];   // [800000]
    const float* W1       = (const float*)d_in[3];   // [256,128]
    const float* b1       = (const float*)d_in[4];
    const float* W2       = (const float*)d_in[5];   // [256,256]
    const float* b2       = (const float*)d_in[6];
    const float* g1       = (const float*)d_in[7];
    const float* beta1    = (const float*)d_in[8];
    const float* g2       = (const float*)d_in[9];
    const float* beta2    = (const float*)d_in[10];
    const float* prelu_a  = (const float*)d_in[11];
    // d_in[12] = num_hyperedges (fixed 20000, baked into constants)

    const size_t nodeRowBytes = (size_t)N_NODES * D * sizeof(float);   // 102.4 MB
    const size_t edgeRowBytes = (size_t)N_EDGES * D * sizeof(float);   //  20.5 MB

    float* B0  = (float*)d_out;                        // node-sized buffer A (also final out)
    char*  ws  = (char*)d_ws;
    float* B1  = (float*)ws;                           // node-sized buffer B
    float* E   = (float*)(ws + nodeRowBytes);          // [20000,256]
    float* deg = E + (size_t)N_EDGES * D;              // [100000]
    float* cnt = deg + N_NODES;                        // [20000]
    float* W2s = cnt + N_EDGES;                        // [256,256]

    const int mTiles = N_NODES / 16;                   // 6250
    const int gemmBlocks = (mTiles * 4 * 32) / 256;    // 3125 blocks of 8 waves

    // degree / cardinality (shared by both conv layers)
    hipMemsetAsync(deg, 0, (size_t)(N_NODES + N_EDGES) * sizeof(float), stream);
    degcnt_kernel<<<(NNZ_CNT + 255) / 256, 256, 0, stream>>>(node_idx, edge_idx, deg, cnt);

    // ---- layer 1 ----
    gemm_wmma_kernel<D_IN><<<gemmBlocks, 256, 0, stream>>>(edge_fea, W1, B1, mTiles);

    hipMemsetAsync(E, 0, edgeRowBytes, stream);
    scatter_rows_kernel<<<NNZ_CNT, 256, 0, stream>>>(B1, node_idx, edge_idx, E);
    scale_edges_kernel<<<(N_EDGES * D) / 256, 256, 0, stream>>>(E, cnt);

    hipMemsetAsync(B0, 0, nodeRowBytes, stream);
    scatter_rows_kernel<<<NNZ_CNT, 256, 0, stream>>>(E, edge_idx, node_idx, B0);
    fused_scale_ln_prelu_kernel<<<N_NODES, 256, 0, stream>>>(B0, deg, b1, g1, beta1, prelu_a, B1);

    // ---- layer 2 ----
    wstd_kernel<<<D, 256, 0, stream>>>(W2, W2s);
    gemm_wmma_kernel<D><<<gemmBlocks, 256, 0, stream>>>(B1, W2s, B0, mTiles);

    hipMemsetAsync(E, 0, edgeRowBytes, stream);
    scatter_rows_kernel<<<NNZ_CNT, 256, 0, stream>>>(B0, node_idx, edge_idx, E);
    scale_edges_kernel<<<(N_EDGES * D) / 256, 256, 0, stream>>>(E, cnt);

    hipMemsetAsync(B1, 0, nodeRowBytes, stream);
    scatter_rows_kernel<<<NNZ_CNT, 256, 0, stream>>>(E, edge_idx, node_idx, B1);
    fused_scale_ln_prelu_kernel<<<N_NODES, 256, 0, stream>>>(B1, deg, b2, g2, beta2, prelu_a, B0);
}